// SurfaceConstructor_38474317038067
// MI455X (gfx1250) — compile-verified
//
#include <hip/hip_runtime.h>
#include <math.h>

namespace {

constexpr int B_ = 4;
constexpr int N_ = 8192;   // centers per batch
constexpr int M_ = 8192;   // context points per batch

typedef __attribute__((ext_vector_type(2))) float v2f;
typedef __attribute__((ext_vector_type(8))) float v8f;

// ---------------------------------------------------------------------------
// Pack context into (x, y, z, |p|^2) float4, contiguous per batch.
// ---------------------------------------------------------------------------
__global__ void prep_ctx(const float* __restrict__ context, float4* __restrict__ ctx4) {
  int t = blockIdx.x * blockDim.x + threadIdx.x;
  if (t >= B_ * M_) return;
  int b = t / M_, j = t - b * M_;
  const float* p = context + (size_t)b * 3 * M_;
  float x = p[j], y = p[M_ + j], z = p[2 * M_ + j];
  ctx4[t] = make_float4(x, y, z, x * x + y * y + z * z);
}

// All-ones mask iff x < y.  (x-y) is exact for nearby floats, sign bit == (x<y).
__device__ __forceinline__ int ltmask(float x, float y) {
  return __float_as_int(x - y) >> 31;
}
// (a & m) | (b & ~m)  -> single v_bfi_b32
__device__ __forceinline__ int blend(int m, int a, int b) {
  return (a & m) | (b & ~m);
}

// Branchless insert of (d, j) into ascending top-3 (d0<=d1<=d2).
// Distances via pure min/max algebra; indices via sign-mask + bfi blends.
__device__ __forceinline__ void insert3(float d, int j,
    float& d0, float& d1, float& d2, int& i0, int& i1, int& i2) {
  const int m0 = ltmask(d, d0);          // d < d0
  const int m1 = ltmask(d, d1);          // d < d1
  const int m2 = ltmask(d, d2);          // d < d2
  const float nd0 = fminf(d, d0);
  const float nd1 = fminf(d1, fmaxf(d, d0));
  const float nd2 = fminf(d2, fmaxf(d, d1));
  const int t01 = blend(m0, i0, j);      // what shifts into slot1 if d<d1
  const int t12 = blend(m1, i1, j);      // what shifts into slot2 if d<d2
  i0 = blend(m0, j, i0);
  i1 = blend(m1, t01, i1);
  i2 = blend(m2, t12, i2);
  d0 = nd0; d1 = nd1; d2 = nd2;
}

// ---------------------------------------------------------------------------
// One wave per 16-row tile.  Distance ranking via V_WMMA_F32_16X16X4_F32:
//   A (16x4) row i   = (-2x_i, -2y_i, -2z_i, 1)
//   B (4x16) col j   = ( x_j,   y_j,   z_j,  |ctx_j|^2)
//   D[i][j] = |ctx_j|^2 - 2 c_i . ctx_j   (== d_ij - |c_i|^2, same row ranking)
// A layout (f32 16x4): VGPR0 = lane<16 ? K0 : K2 ; VGPR1 = lane<16 ? K1 : K3.
// B layout mirrors it.  C/D: VGPR r, lane l -> row r + 8*(l>=16), col l%16.
// ---------------------------------------------------------------------------
__global__ __launch_bounds__(256) void knn_surface(
    const float* __restrict__ center, const float4* __restrict__ ctx4,
    float* __restrict__ out, int* __restrict__ flags)
{
  const int lane = threadIdx.x & 31;
  const int wid  = threadIdx.x >> 5;
  const int gw   = blockIdx.x * 8 + wid;        // global wave id, 0..2047
  const int b    = gw >> 9;                     // / (N_/16)
  const int rowbase = (gw & 511) << 4;
  const int col  = lane & 15;
  const bool hi  = lane >= 16;

  // Build A operand (held in 2 VGPRs for the whole sweep).
  const float* cb = center + (size_t)b * 3 * N_;
  const int mrow = rowbase + col;
  const float ax = -2.0f * cb[mrow];
  const float ay = -2.0f * cb[N_ + mrow];
  const float az = -2.0f * cb[2 * N_ + mrow];
  v2f A;
  A[0] = hi ? az : ax;     // K = lane<16 ? 0 : 2
  A[1] = hi ? 1.0f : ay;   // K = lane<16 ? 1 : 3

  const float4* cx = ctx4 + (size_t)b * M_;

  // Per-lane running top-3 for 8 rows (this lane's column subset).
  float d0[8], d1[8], d2[8];
  int   i0[8], i1[8], i2[8];
#pragma unroll
  for (int r = 0; r < 8; ++r) { d0[r] = d1[r] = d2[r] = 3.4e38f; i0[r] = i1[r] = i2[r] = 0; }

  const v8f zero = {};
  constexpr int NT = M_ / 16;

#define PROCESS_TILE(TILE, PVEC)                                               \
  do {                                                                         \
    const int j_ = ((TILE) << 4) + col;                                        \
    v2f Bm_;                                                                   \
    Bm_[0] = hi ? (PVEC).z : (PVEC).x;  /* K0 / K2 */                          \
    Bm_[1] = hi ? (PVEC).w : (PVEC).y;  /* K1 / K3 (|ctx|^2) */                \
    v8f acc_ = __builtin_amdgcn_wmma_f32_16x16x4_f32(                          \
        false, A, false, Bm_, (short)0, zero, false, false);                   \
    _Pragma("unroll")                                                          \
    for (int r = 0; r < 8; ++r)                                                \
      insert3(acc_[r], j_, d0[r], d1[r], d2[r], i0[r], i1[r], i2[r]);          \
  } while (0)

  // Software-pipelined sweep over M: prefetch tile t+1 while inserting tile t.
  float4 p = cx[col];
  for (int tile = 0; tile < NT - 1; ++tile) {
    float4 pn = cx[((tile + 1) << 4) + col];   // global_load_b128, L2-resident
    PROCESS_TILE(tile, p);
    p = pn;
  }
  PROCESS_TILE(NT - 1, p);
#undef PROCESS_TILE

  // Merge partial top-3 lists across the 16 lanes of each half-wave.
#pragma unroll
  for (int mask = 1; mask < 16; mask <<= 1) {
#pragma unroll
    for (int r = 0; r < 8; ++r) {
      float od0 = __shfl_xor(d0[r], mask, 32); int oi0 = __shfl_xor(i0[r], mask, 32);
      float od1 = __shfl_xor(d1[r], mask, 32); int oi1 = __shfl_xor(i1[r], mask, 32);
      float od2 = __shfl_xor(d2[r], mask, 32); int oi2 = __shfl_xor(i2[r], mask, 32);
      insert3(od0, oi0, d0[r], d1[r], d2[r], i0[r], i1[r], i2[r]);
      insert3(od1, oi1, d0[r], d1[r], d2[r], i0[r], i1[r], i2[r]);
      insert3(od2, oi2, d0[r], d1[r], d2[r], i0[r], i1[r], i2[r]);
    }
  }

  // Geometry: lane 0 handles rows rowbase+0..7, lane 16 handles rowbase+8..15.
  if (col == 0) {
    float* out_unit = out;
    float* out_cen  = out + (size_t)B_ * 3 * N_;
    float* out_pos  = out + (size_t)2 * B_ * 3 * N_;
    const int half = hi ? 8 : 0;
#pragma unroll
    for (int r = 0; r < 8; ++r) {
      const int row = rowbase + r + half;
      float4 p0 = cx[i0[r]], p1 = cx[i1[r]], p2 = cx[i2[r]];
      float e1x = p1.x - p0.x, e1y = p1.y - p0.y, e1z = p1.z - p0.z;
      float e2x = p2.x - p0.x, e2y = p2.y - p0.y, e2z = p2.z - p0.z;
      float nx = e1y * e2z - e1z * e2y;
      float ny = e1z * e2x - e1x * e2z;
      float nz = e1x * e2y - e1y * e2x;
      float nn = sqrtf(nx * nx + ny * ny + nz * nz);
      float ux = nx / nn, uy = ny / nn, uz = nz / nn;   // NaN when degenerate
      float s = (ux > 0.0f) ? 1.0f : -1.0f;
      ux *= s; uy *= s; uz *= s;
      float gx = (p0.x + p1.x + p2.x) * (1.0f / 3.0f);
      float gy = (p0.y + p1.y + p2.y) * (1.0f / 3.0f);
      float gz = (p0.z + p1.z + p2.z) * (1.0f / 3.0f);
      float pv = (ux * gx + uy * gy + uz * gz) * 0.57735026918962576f; // 1/sqrt(3)
      size_t base = (size_t)b * 3 * N_;
      out_unit[base + row]          = ux;
      out_unit[base + N_ + row]     = uy;
      out_unit[base + 2 * N_ + row] = uz;
      out_cen[base + row]           = gx;
      out_cen[base + N_ + row]      = gy;
      out_cen[base + 2 * N_ + row]  = gz;
      out_pos[(size_t)b * N_ + row] = pv;
      flags[b * N_ + row] = (ux != ux || uy != uy || uz != uz) ? 1 : 0;
    }
  }
}

// ---------------------------------------------------------------------------
// Per-batch NaN fixup: replace flagged rows with first unflagged row.
// ---------------------------------------------------------------------------
__global__ void nan_fixup(float* __restrict__ out, const int* __restrict__ flags) {
  __shared__ int s_first;
  const int b = blockIdx.x;
  const int* f = flags + b * N_;
  if (threadIdx.x == 0) s_first = N_;
  __syncthreads();
  int local = N_;
  for (int n = threadIdx.x; n < N_; n += blockDim.x)
    if (f[n] == 0) { local = n; break; }   // strided scan: first hit is the min
  atomicMin(&s_first, local);
  __syncthreads();
  const int first = (s_first == N_) ? 0 : s_first;   // argmax(~mask) semantics

  float* out_unit = out;
  float* out_cen  = out + (size_t)B_ * 3 * N_;
  float* out_pos  = out + (size_t)2 * B_ * 3 * N_;
  size_t base = (size_t)b * 3 * N_;
  float su0 = out_unit[base + first], su1 = out_unit[base + N_ + first], su2 = out_unit[base + 2 * N_ + first];
  float sc0 = out_cen[base + first],  sc1 = out_cen[base + N_ + first],  sc2 = out_cen[base + 2 * N_ + first];
  float sp  = out_pos[(size_t)b * N_ + first];
  __syncthreads();
  for (int n = threadIdx.x; n < N_; n += blockDim.x) {
    if (f[n]) {
      out_unit[base + n] = su0; out_unit[base + N_ + n] = su1; out_unit[base + 2 * N_ + n] = su2;
      out_cen[base + n]  = sc0; out_cen[base + N_ + n]  = sc1; out_cen[base + 2 * N_ + n]  = sc2;
      out_pos[(size_t)b * N_ + n] = sp;
    }
  }
}

} // namespace

extern "C" void kernel_launch(void* const* d_in, const int* in_sizes, int n_in,
                              void* d_out, int out_size, void* d_ws, size_t ws_size,
                              hipStream_t stream) {
  const float* center  = (const float*)d_in[0];   // [B,3,N]
  const float* context = (const float*)d_in[1];   // [B,3,M]
  float4* ctx4 = (float4*)d_ws;                               // B*M float4 = 512 KB
  int* flags   = (int*)((char*)d_ws + (size_t)B_ * M_ * sizeof(float4)); // B*N ints
  float* out   = (float*)d_out;                   // unit | cen | pos, concatenated

  const int tot = B_ * M_;
  prep_ctx<<<(tot + 255) / 256, 256, 0, stream>>>(context, ctx4);
  knn_surface<<<(B_ * (N_ / 16)) / 8, 256, 0, stream>>>(center, ctx4, out, flags);
  nan_fixup<<<B_, 256, 0, stream>>>(out, flags);
}